// LindBladEvolve_24524263260255
// MI455X (gfx1250) — compile-verified
//
#include <hip/hip_runtime.h>
#include <stdint.h>

// ---------------------------------------------------------------------------
// Monte-Carlo Lindblad trajectory scan, MI455X (gfx1250, wave32).
//  - 1 lane = 1 trajectory, sequential loop over T (scan: no T-parallelism).
//  - Broadcast tables (P, H) DMA'd into LDS via TDM tensor_load_to_lds,
//    including a strided-gather descriptor for H_im[t,0,1].
//  - Per-step uniform ds_load_b128 broadcasts, software-pipelined by 1 step.
//  - Non-temporal loads + distance-24 global_prefetch for the 64MB r/u
//    streams; wave-uniform rare-jump branch via __any().
// ---------------------------------------------------------------------------

#define MAX_T 1024

typedef unsigned int u32x4 __attribute__((ext_vector_type(4)));
typedef int          i32x8 __attribute__((ext_vector_type(8)));
typedef int          i32x4 __attribute__((ext_vector_type(4)));

static __device__ __forceinline__ uint32_t lds_byte_off(const void* p) {
    // generic pointer to __shared__: low 32 bits are the LDS byte address
    return (uint32_t)(uintptr_t)p;
}

#if defined(__has_builtin)
#if __has_builtin(__builtin_amdgcn_tensor_load_to_lds)
#define HAVE_TDM 1
#endif
#endif

// 2D tile load: td1 rows of td0 float32 elements, row stride stride0 elements,
// packed contiguously into LDS at lds_byte.
static __device__ __forceinline__ void tdm_load_2d(const float* gsrc, uint32_t lds_byte,
                                                   uint32_t td0, uint32_t td1,
                                                   uint32_t stride0) {
#if defined(HAVE_TDM)
    uint64_t ga = (uint64_t)(uintptr_t)gsrc;
    u32x4 g0;
    g0[0] = 1u;                                   // count=1, user descriptor
    g0[1] = lds_byte;                             // lds_addr [63:32]
    g0[2] = (uint32_t)ga;                         // global_addr [95:64]
    g0[3] = ((uint32_t)(ga >> 32) & 0x01FFFFFFu)  // global_addr [120:96]
          | (2u << 30);                           // type=2 ("image") [127:126]
    i32x8 g1;
    g1[0] = (int)(2u << 16);                      // wg_mask=0, data_size=4B
    g1[1] = (int)((td0 & 0xFFFFu) << 16);         // tensor_dim0[15:0] @ bit48
    g1[2] = (int)(((td0 >> 16) & 0xFFFFu) | ((td1 & 0xFFFFu) << 16)); // dim0 hi | dim1 lo
    g1[3] = (int)(((td1 >> 16) & 0xFFFFu) | ((td0 & 0xFFFFu) << 16)); // dim1 hi | tile_dim0
    g1[4] = (int)(td1 & 0xFFFFu);                 // tile_dim1 | tile_dim2=0
    g1[5] = (int)stride0;                         // tensor_dim0_stride lo
    g1[6] = 0;                                    // stride0 hi | dim1_stride lo
    g1[7] = 0;
    i32x4 z4; z4[0] = z4[1] = z4[2] = z4[3] = 0;  // groups 2/3 unused (<=2D)
    i32x8 z8; z8[0] = z8[1] = z8[2] = z8[3] = z8[4] = z8[5] = z8[6] = z8[7] = 0;
    __builtin_amdgcn_tensor_load_to_lds(g0, g1, z4, z4, z8, 0);
#else
    (void)gsrc; (void)lds_byte; (void)td0; (void)td1; (void)stride0;
#endif
}

__global__ __launch_bounds__(256)
void lindblad_scan_kernel(const float* __restrict__ H_re, const float* __restrict__ H_im,
                          const float* __restrict__ P_re, const float* __restrict__ P_im,
                          const float* __restrict__ C_re, const float* __restrict__ C_im,
                          const float* __restrict__ r0,   const float* __restrict__ r_stream,
                          const float* __restrict__ u_jump, const int* __restrict__ init_idx,
                          float* __restrict__ out, int Tn, int Bn)
{
    __shared__ __align__(16) float sPre[MAX_T * 4];   // P real, [t][2][2]
    __shared__ __align__(16) float sPim[MAX_T * 4];   // P imag
    __shared__ __align__(16) float sHre[MAX_T * 4];   // H real
    __shared__ __align__(16) float sHim1[MAX_T];      // H_im[t,0,1] (gathered, stride 4)

#if defined(HAVE_TDM)
    if (threadIdx.x < 32) {   // one wave issues the 4 DMA descriptors
        tdm_load_2d(P_re,     lds_byte_off(sPre),  (uint32_t)(Tn * 4), 1u, (uint32_t)(Tn * 4));
        tdm_load_2d(P_im,     lds_byte_off(sPim),  (uint32_t)(Tn * 4), 1u, (uint32_t)(Tn * 4));
        tdm_load_2d(H_re,     lds_byte_off(sHre),  (uint32_t)(Tn * 4), 1u, (uint32_t)(Tn * 4));
        tdm_load_2d(H_im + 1, lds_byte_off(sHim1), 1u, (uint32_t)Tn, 4u); // strided gather
        __builtin_amdgcn_s_wait_tensorcnt(0);
    }
#else
    for (int i = threadIdx.x; i < Tn * 4; i += blockDim.x) {
        sPre[i] = P_re[i]; sPim[i] = P_im[i]; sHre[i] = H_re[i];
    }
    for (int i = threadIdx.x; i < Tn; i += blockDim.x) sHim1[i] = H_im[4 * i + 1];
#endif
    __syncthreads();

    int b = blockIdx.x * blockDim.x + threadIdx.x;
    if (b >= Bn) return;

    // ---- precompute CdC_k = C_k^dag C_k (Hermitian: m00, m11 real; m01 complex), K=2
    float c0[8], c1[8];   // [r00,i00, r01,i01, r10,i10, r11,i11]
#pragma unroll
    for (int e = 0; e < 4; ++e) {
        c0[2 * e] = C_re[e];     c0[2 * e + 1] = C_im[e];
        c1[2 * e] = C_re[4 + e]; c1[2 * e + 1] = C_im[4 + e];
    }
    float m00_0 = c0[0]*c0[0] + c0[1]*c0[1] + c0[4]*c0[4] + c0[5]*c0[5];
    float m11_0 = c0[2]*c0[2] + c0[3]*c0[3] + c0[6]*c0[6] + c0[7]*c0[7];
    float m01r0 = c0[0]*c0[2] + c0[1]*c0[3] + c0[4]*c0[6] + c0[5]*c0[7];
    float m01i0 = c0[0]*c0[3] - c0[1]*c0[2] + c0[4]*c0[7] - c0[5]*c0[6];
    float m00_1 = c1[0]*c1[0] + c1[1]*c1[1] + c1[4]*c1[4] + c1[5]*c1[5];
    float m11_1 = c1[2]*c1[2] + c1[3]*c1[3] + c1[6]*c1[6] + c1[7]*c1[7];
    float m01r1 = c1[0]*c1[2] + c1[1]*c1[3] + c1[4]*c1[6] + c1[5]*c1[7];
    float m01i1 = c1[0]*c1[3] - c1[1]*c1[2] + c1[4]*c1[7] - c1[5]*c1[6];

    // ---- per-trajectory state
    int ii = init_idx[b];
    float p0r = (ii == 0) ? 1.f : 0.f, p0i = 0.f;
    float p1r = (ii == 0) ? 0.f : 1.f, p1i = 0.f;
    float rr   = r0[b];
    float prob = 1.f;
    float en   = 0.f;

    const float* rp = r_stream + b;
    const float* up = u_jump + b;

    // ---- software pipeline: step-(t) consumes values loaded at step-(t-1)
    float rs_n = __builtin_nontemporal_load(rp);
    float u_n  = __builtin_nontemporal_load(up);
    float4 Pr_n = *(const float4*)(sPre);
    float4 Pi_n = *(const float4*)(sPim);
    float4 Hr_n = *(const float4*)(sHre);
    float  Hi_n = sHim1[0];

    for (int t = 0; t < Tn; ++t) {
        float  rs_t = rs_n, u_t = u_n;
        float4 Pr = Pr_n, Pi = Pi_n, Hr = Hr_n;
        float  H01i = Hi_n;

        int t1 = (t + 1 < Tn) ? (t + 1) : t;
        rs_n = __builtin_nontemporal_load(rp + (size_t)t1 * Bn);
        u_n  = __builtin_nontemporal_load(up + (size_t)t1 * Bn);
        Pr_n = *(const float4*)(sPre + 4 * t1);
        Pi_n = *(const float4*)(sPim + 4 * t1);
        Hr_n = *(const float4*)(sHre + 4 * t1);
        Hi_n = sHim1[t1];

        int tp = (t + 24 < Tn) ? (t + 24) : t;       // stream prefetch distance
        __builtin_prefetch(rp + (size_t)tp * Bn, 0, 0);
        __builtin_prefetch(up + (size_t)tp * Bn, 0, 0);

        // cand = P_t @ psi  (complex 2x2 matvec)
        float c0r = Pr.x*p0r - Pi.x*p0i + Pr.y*p1r - Pi.y*p1i;
        float c0i = Pr.x*p0i + Pi.x*p0r + Pr.y*p1i + Pi.y*p1r;
        float c1r = Pr.z*p0r - Pi.z*p0i + Pr.w*p1r - Pi.w*p1i;
        float c1i = Pr.z*p0i + Pi.z*p0r + Pr.w*p1i + Pi.w*p1r;
        float n2  = c0r*c0r + c0i*c0i + c1r*c1r + c1i*c1i;

        bool jump = (n2 <= rr) || (n2 < 1e-9f);

        float np0r = c0r, np0i = c0i, np1r = c1r, np1i = c1i;
        if (__any(jump)) {                            // rare path, wave-uniform skip
            float ap0 = p0r*p0r + p0i*p0i;
            float ap1 = p1r*p1r + p1i*p1i;
            float on2 = ap0 + ap1;
            float ar  = p0r*p1r + p0i*p1i;            // Re(conj(p0)*p1)
            float ai  = p0r*p1i - p0i*p1r;            // Im(conj(p0)*p1)
            float jp0 = m00_0*ap0 + m11_0*ap1 + 2.f*(m01r0*ar - m01i0*ai);
            float jp1 = m00_1*ap0 + m11_1*ap1 + 2.f*(m01r1*ar - m01i1*ai);
            float inv = 1.f / (jp0 + jp1);
            float cdf0 = jp0 * inv;
            bool  sel1 = (u_t >= cdf0);               // channel index (K=2, clipped)
            float psel = sel1 ? (jp1 * inv) : cdf0;
            // jpsi = C[idx] @ psi
            float k00r = sel1 ? c1[0] : c0[0], k00i = sel1 ? c1[1] : c0[1];
            float k01r = sel1 ? c1[2] : c0[2], k01i = sel1 ? c1[3] : c0[3];
            float k10r = sel1 ? c1[4] : c0[4], k10i = sel1 ? c1[5] : c0[5];
            float k11r = sel1 ? c1[6] : c0[6], k11i = sel1 ? c1[7] : c0[7];
            float j0r = k00r*p0r - k00i*p0i + k01r*p1r - k01i*p1i;
            float j0i = k00r*p0i + k00i*p0r + k01r*p1i + k01i*p1r;
            float j1r = k10r*p0r - k10i*p0i + k11r*p1r - k11i*p1i;
            float j1i = k10r*p0i + k10i*p0r + k11r*p1i + k11i*p1r;
            float jn2 = fmaxf(j0r*j0r + j0i*j0i + j1r*j1r + j1i*j1i, 1e-20f);
            float rin = rsqrtf(jn2);
            j0r *= rin; j0i *= rin; j1r *= rin; j1i *= rin;

            np0r = jump ? j0r : c0r;  np0i = jump ? j0i : c0i;
            np1r = jump ? j1r : c1r;  np1i = jump ? j1i : c1i;
            prob = jump ? prob * on2 * psel : prob;
            rr   = jump ? rs_t : rr;
        }
        p0r = np0r; p0i = np0i; p1r = np1r; p1i = np1i;

        // energy: Re(psi^dag H psi), H Hermitian
        float br = p0r*p1r + p0i*p1i;
        float bi = p0r*p1i - p0i*p1r;
        float e  = Hr.x*(p0r*p0r + p0i*p0i) + Hr.w*(p1r*p1r + p1i*p1i)
                 + 2.f*(Hr.y*br - H01i*bi);
        en += e;
    }

    float nf = p0r*p0r + p0i*p0i + p1r*p1r + p1i*p1i;
    __builtin_nontemporal_store(en,        out + b);
    __builtin_nontemporal_store(prob * nf, out + Bn + b);
}

extern "C" void kernel_launch(void* const* d_in, const int* in_sizes, int n_in,
                              void* d_out, int out_size, void* d_ws, size_t ws_size,
                              hipStream_t stream) {
    const float* H_re     = (const float*)d_in[0];
    const float* H_im     = (const float*)d_in[1];
    const float* P_re     = (const float*)d_in[2];
    const float* P_im     = (const float*)d_in[3];
    const float* C_re     = (const float*)d_in[4];
    const float* C_im     = (const float*)d_in[5];
    const float* r0       = (const float*)d_in[6];
    const float* r_stream = (const float*)d_in[7];
    const float* u_jump   = (const float*)d_in[8];
    const int*   init_idx = (const int*)d_in[9];
    float* out = (float*)d_out;

    int B = in_sizes[6];           // r0 length
    int T = in_sizes[0] / 4;       // H_re is [T,2,2]
    if (T > MAX_T) T = MAX_T;

    int threads = 256;             // 8 wave32 per workgroup
    int blocks  = (B + threads - 1) / threads;
    hipLaunchKernelGGL(lindblad_scan_kernel, dim3(blocks), dim3(threads), 0, stream,
                       H_re, H_im, P_re, P_im, C_re, C_im, r0, r_stream, u_jump,
                       init_idx, out, T, B);
}